// TrainablePCEN_41308995453743
// MI455X (gfx1250) — compile-verified
//
#include <hip/hip_runtime.h>
#include <math.h>

// PCEN on MI455X (gfx1250, wave32).
// EMA scan as 16x16 tiles: P = E x L via 4x v_wmma_f32_16x16x4_f32 per tile
// (two independent accumulator chains), rank-1 carry update in VALU,
// ds_bpermute carry broadcast.
// Input streamed with double-buffered global_load_async_to_lds_b128 (ASYNCcnt).
// All powers in base-2 via raw v_log_f32 / v_exp_f32 (operands provably normal).

typedef float v2f __attribute__((ext_vector_type(2)));
typedef float v8f __attribute__((ext_vector_type(8)));

#define T_LEN          8000
#define N_ROWS         8192       // 64 * 128
#define ROWS_PER_TILE  16
#define N_TILES        (N_ROWS / ROWS_PER_TILE)   // 512
#define N_SEG          5
#define CH_PER_SEG     50         // 50 * 32 = 1600 timesteps per segment
#define WARM_CH        16         // 512-step warmup; a^512 ~ 2.4e-6
#define WAVES_PER_BLK  8
#define LDS_STRIDE     36         // 32 cols + pad -> conflict-free b64/b32 reads
#define TILE_FLOATS    (ROWS_PER_TILE * LDS_STRIDE)   // 576
#define TILE_BYTES     (TILE_FLOATS * 4)              // 2304

#define S_COEF  0.025f
#define A_COEF  0.975f
#define EPS_C   1e-6f
#define LOG2E   1.4426950408889634f

// Issue 4 async b128 loads: 16 rows x 32 t (128B/row) -> LDS tile (raw x).
__device__ __forceinline__ void async_load_tile(const float* gbase,
                                                unsigned ldsbase, int lane)
{
    #pragma unroll
    for (int i = 0; i < 4; ++i) {
        const int q  = lane + 32 * i;
        const int rw = q >> 3;
        const int c4 = (q & 7) << 2;
        const unsigned long long ga =
            (unsigned long long)(uintptr_t)(gbase + rw * T_LEN + c4);
        const unsigned lo = ldsbase + (unsigned)((rw * LDS_STRIDE + c4) * 4);
        asm volatile("global_load_async_to_lds_b128 %0, %1, off"
                     :: "v"(lo), "v"(ga) : "memory");
    }
}

__launch_bounds__(256)
__global__ void pcen_wmma_kernel(const float* __restrict__ x,
                                 const float* __restrict__ p_la,
                                 const float* __restrict__ p_ld,
                                 const float* __restrict__ p_lr,
                                 float* __restrict__ out)
{
    __shared__ float lds[WAVES_PER_BLK][2][TILE_FLOATS];

    const int lane = threadIdx.x & 31;
    const int wib  = threadIdx.x >> 5;
    // Wave-uniform control state -> SGPRs -> scalar branches; EXEC stays
    // all-ones around WMMA / ds_bpermute (ISA requirement).
    const int wg   = __builtin_amdgcn_readfirstlane(blockIdx.x * WAVES_PER_BLK + wib);
    const int tile = wg / N_SEG;
    const int seg  = wg % N_SEG;
    if (tile >= N_TILES) return;

    float*         tw  = &lds[wib][0][0];
    const unsigned lwb = (unsigned)(uintptr_t)tw;   // LDS byte offset of wave's buffers

    // Runtime scalars (uniform loads); base-2 domain throughout.
    const float alpha = fminf(fmaxf(__builtin_amdgcn_exp2f(p_la[0] * LOG2E), 0.01f), 1.0f);
    const float delta = fminf(fmaxf(__builtin_amdgcn_exp2f(p_ld[0] * LOG2E), 0.01f), 5.0f);
    const float rr_   = fminf(fmaxf(__builtin_amdgcn_exp2f(p_lr[0] * LOG2E), 0.01f), 1.0f);
    const float dr    = __builtin_amdgcn_exp2f(rr_ * __builtin_amdgcn_logf(delta));

    const int   cc   = lane & 15;                    // tile column (N index)
    const int   hh   = lane >> 4;                    // half-wave
    const float l2a  = __builtin_amdgcn_logf(A_COEF);                  // log2(0.975)
    const float apc1 = __builtin_amdgcn_exp2f(l2a * (float)(cc + 1));  // a^(c+1)

    // Constant B fragments: L[j][n] = (j<=n) ? S*a^(n-j) : 0, fragment k = rows 4k..4k+3.
    // B 4x16 layout (mirrors A table): VGPR0 = row (4k+2h), VGPR1 = row (4k+2h+1).
    v2f bf[4];
    #pragma unroll
    for (int k = 0; k < 4; ++k) {
        const int j0 = 4 * k + 2 * hh;
        bf[k].x = (j0     <= cc) ? S_COEF * __builtin_amdgcn_exp2f(l2a * (float)(cc - j0))     : 0.0f;
        bf[k].y = (j0 + 1 <= cc) ? S_COEF * __builtin_amdgcn_exp2f(l2a * (float)(cc - j0 - 1)) : 0.0f;
    }

    const float* __restrict__ xb = x   + (size_t)tile * ROWS_PER_TILE * T_LEN;
    float*       __restrict__ ob = out + (size_t)tile * ROWS_PER_TILE * T_LEN;

    const int ch_out0 = seg * CH_PER_SEG;
    const int ch0     = ch_out0 - (seg ? WARM_CH : 0);
    const int ch1     = ch_out0 + CH_PER_SEG;

    float carry[8];
    #pragma unroll
    for (int v = 0; v < 8; ++v) carry[v] = 0.0f;

    const int bperm_idx = ((lane & 16) | 15) << 2;   // lane 15 / 31 broadcast

    // Prologue: start streaming the first chunk.
    async_load_tile(xb + (size_t)ch0 * 32, lwb + (unsigned)((ch0 & 1) * TILE_BYTES), lane);

    for (int ch = ch0; ch < ch1; ++ch) {
        const int  tb   = ch * 32;
        const bool emit = (ch >= ch_out0);
        float* __restrict__ tl = tw + (ch & 1) * TILE_FLOATS;

        if (ch + 1 < ch1) {
            // WAR guard: previous chunk's LDS reads of the other buffer must retire.
            asm volatile("s_wait_dscnt 0x0" ::: "memory");
            async_load_tile(xb + (size_t)(ch + 1) * 32,
                            lwb + (unsigned)(((ch + 1) & 1) * TILE_BYTES), lane);
            // Async loads complete in order: <=4 outstanding => current buffer landed.
            asm volatile("s_wait_asynccnt 0x4" ::: "memory");
        } else {
            asm volatile("s_wait_asynccnt 0x0" ::: "memory");
        }
        __builtin_amdgcn_wave_barrier();

        #pragma unroll
        for (int h2 = 0; h2 < 2; ++h2) {
            const int c0 = 16 * h2;

            // P = E x L via 4 f32 WMMAs (K=4 each) in two independent chains.
            v8f acc01 = {0.f, 0.f, 0.f, 0.f, 0.f, 0.f, 0.f, 0.f};
            v8f acc23 = {0.f, 0.f, 0.f, 0.f, 0.f, 0.f, 0.f, 0.f};
            #pragma unroll
            for (int k = 0; k < 4; ++k) {
                // A 16x4 f32: lane holds row (lane%16), K = {2h, 2h+1} of sub-block.
                const v2f raw = *(const v2f*)&tl[cc * LDS_STRIDE + c0 + 4 * k + 2 * hh];
                v2f a;
                a.x = __builtin_amdgcn_exp2f(raw.x * LOG2E);
                a.y = __builtin_amdgcn_exp2f(raw.y * LOG2E);
                if (k < 2)
                    acc01 = __builtin_amdgcn_wmma_f32_16x16x4_f32(
                                false, a, false, bf[k], (short)0, acc01, false, false);
                else
                    acc23 = __builtin_amdgcn_wmma_f32_16x16x4_f32(
                                false, a, false, bf[k], (short)0, acc23, false, false);
            }

            // Carry-in + carry-out broadcast (always, including warmup chunks).
            float marr[8];
            #pragma unroll
            for (int v = 0; v < 8; ++v) {
                const float m = acc01[v] + __builtin_fmaf(apc1, carry[v], acc23[v]);
                marr[v] = m;
                carry[v] = __uint_as_float((unsigned)__builtin_amdgcn_ds_bpermute(
                               bperm_idx, (int)__float_as_uint(m)));
            }

            if (emit) {   // single scalar branch per half
                // D layout: lane holds row v + 8*hh, column cc. Batch the 8 e-loads,
                // then 8 independent 3-trans chains for ILP.
                float ev[8];
                #pragma unroll
                for (int v = 0; v < 8; ++v)
                    ev[v] = tl[(v + 8 * hh) * LDS_STRIDE + c0 + cc];
                #pragma unroll
                for (int v = 0; v < 8; ++v) {
                    const float e    = __builtin_amdgcn_exp2f(ev[v] * LOG2E);
                    const float lm   = __builtin_amdgcn_logf(marr[v] + EPS_C);
                    const float invp = __builtin_amdgcn_exp2f(-alpha * lm);
                    const float u    = __builtin_fmaf(e, invp, delta);
                    tl[(v + 8 * hh) * LDS_STRIDE + c0 + cc] =
                        __builtin_amdgcn_exp2f(rr_ * __builtin_amdgcn_logf(u)) - dr;
                }
            }
        }
        __builtin_amdgcn_wave_barrier();

        if (emit) {
            // Coalesced 128B/row stores from LDS.
            #pragma unroll
            for (int i = 0; i < 4; ++i) {
                const int q  = lane + 32 * i;
                const int rw = q >> 3;
                const int c4 = (q & 7) << 2;
                const float4 o4 = *(const float4*)&tl[rw * LDS_STRIDE + c4];
                *(float4*)(ob + rw * T_LEN + tb + c4) = o4;
            }
        }
        __builtin_amdgcn_wave_barrier();
    }
}

extern "C" void kernel_launch(void* const* d_in, const int* in_sizes, int n_in,
                              void* d_out, int out_size, void* d_ws, size_t ws_size,
                              hipStream_t stream) {
    (void)in_sizes; (void)n_in; (void)out_size; (void)d_ws; (void)ws_size;
    const float* x    = (const float*)d_in[0];
    const float* p_la = (const float*)d_in[1];
    const float* p_ld = (const float*)d_in[2];
    const float* p_lr = (const float*)d_in[3];
    float*       out  = (float*)d_out;

    const int total_waves = N_TILES * N_SEG;              // 2560
    const int blocks      = total_waves / WAVES_PER_BLK;  // 320
    pcen_wmma_kernel<<<blocks, WAVES_PER_BLK * 32, 0, stream>>>(x, p_la, p_ld, p_lr, out);
}